// GCNLayer_7000796693164
// MI455X (gfx1250) — compile-verified
//
#include <hip/hip_runtime.h>

typedef float v2f __attribute__((ext_vector_type(2)));
typedef float v4f __attribute__((ext_vector_type(4)));
typedef float v8f __attribute__((ext_vector_type(8)));

#define D 128   // D_IN == D_OUT == 128

// ---------------------------------------------------------------------------
// deg[i] = 1.0  (self-loop weight folded in)
__global__ void k_init_deg(float* __restrict__ deg, int n) {
  int i = blockIdx.x * blockDim.x + threadIdx.x;
  if (i < n) deg[i] = 1.0f;
}

// deg[dst[e]] += ew[e]
__global__ void k_deg_scatter(const long long* __restrict__ ei,
                              const float* __restrict__ ew,
                              float* __restrict__ deg, int E) {
  int e = blockIdx.x * blockDim.x + threadIdx.x;
  if (e < E) {
    long long d = ei[(long long)E + e];
    unsafeAtomicAdd(&deg[d], ew[e]);
  }
}

// dinv[i] = deg>0 ? rsqrt(deg) : 0   (in place)
__global__ void k_dinv(float* __restrict__ deg, int n) {
  int i = blockIdx.x * blockDim.x + threadIdx.x;
  if (i < n) {
    float v = deg[i];
    deg[i] = (v > 0.0f) ? rsqrtf(v) : 0.0f;
  }
}

// Wt[n*D + k] = W[k*D + n]
__global__ void k_transpose(const float* __restrict__ W, float* __restrict__ Wt) {
  int i = blockIdx.x * blockDim.x + threadIdx.x;  // D*D threads
  int k = i >> 7, n = i & (D - 1);
  Wt[n * D + k] = W[i];
}

// ---------------------------------------------------------------------------
// h = x @ W via V_WMMA_F32_16X16X4_F32.
// One wave computes a 16-row tile across all 8 column tiles (full D_OUT=128).
// A fragment (16x4 f32): lanes 0-15 hold M=lane, K=k0+{0,1}; lanes 16-31 hold
// M=lane-16, K=k0+{2,3}  -> contiguous float2 from the x row.
// B fragment (4x16 f32): symmetric, read as contiguous float2 from Wt rows.
__global__ __launch_bounds__(128) void
k_gemm_wmma(const float* __restrict__ x, const float* __restrict__ Wt,
            float* __restrict__ h, int n) {
  const int wave = threadIdx.x >> 5;
  const int lane = threadIdx.x & 31;
  const int half = lane >> 4;      // 0: K low pair, 1: K high pair
  const int m    = lane & 15;

  const int numTiles = (n + 15) >> 4;
  const int tile = blockIdx.x * 4 + wave;     // wave-uniform
  if (tile >= numTiles) return;               // whole wave exits together

  const int row0 = tile * 16;
  int arow = row0 + m;
  if (arow >= n) arow = n - 1;                // clamp (data only, EXEC intact)

  const float* ap = x  + (size_t)arow * D + 2 * half;
  const float* bp = Wt + (size_t)m    * D + 2 * half;

  v8f acc[8];
#pragma unroll
  for (int t = 0; t < 8; ++t) acc[t] = (v8f){0, 0, 0, 0, 0, 0, 0, 0};

#pragma unroll 4
  for (int k0 = 0; k0 < D; k0 += 4) {
    v2f a = *(const v2f*)(ap + k0);
#pragma unroll
    for (int t = 0; t < 8; ++t) {
      v2f b = *(const v2f*)(bp + (size_t)t * 16 * D + k0);
      acc[t] = __builtin_amdgcn_wmma_f32_16x16x4_f32(
          /*neg_a=*/false, a, /*neg_b=*/false, b,
          /*c_mod=*/(short)0, acc[t], /*reuse_a=*/false, /*reuse_b=*/false);
    }
  }

  // C/D layout: VGPR r, lanes 0-15 -> M=r, N=lane; lanes 16-31 -> M=8+r.
#pragma unroll
  for (int t = 0; t < 8; ++t) {
    const int n0 = t * 16;
#pragma unroll
    for (int r = 0; r < 8; ++r) {
      int row = row0 + r + 8 * half;
      if (row < n) h[(size_t)row * D + n0 + m] = acc[t][r];
    }
  }
}

// ---------------------------------------------------------------------------
// out[i,:] = h[i,:] * dinv[i]^2   (self-loop message; also initializes out)
__global__ void k_selfloop(const float* __restrict__ h,
                           const float* __restrict__ dinv,
                           float* __restrict__ out, long long total) {
  long long i = (long long)blockIdx.x * blockDim.x + threadIdx.x;
  if (i < total) {
    float di = dinv[i >> 7];
    out[i] = h[i] * di * di;
  }
}

// one wave per edge: gather 512B row of h[src], scale, fp32-atomic scatter
__global__ __launch_bounds__(256) void
k_edge_scatter(const float* __restrict__ h, const long long* __restrict__ ei,
               const float* __restrict__ ew, const float* __restrict__ dinv,
               float* __restrict__ out, int E) {
  int e = (int)(((long long)blockIdx.x * blockDim.x + threadIdx.x) >> 5);
  int lane = threadIdx.x & 31;
  if (e >= E) return;
  long long s = ei[e];
  long long d = ei[(long long)E + e];
  float nrm = dinv[s] * ew[e] * dinv[d];
  v4f v = ((const v4f*)(h + (size_t)s * D))[lane];
  float* od = out + (size_t)d * D + lane * 4;
  unsafeAtomicAdd(od + 0, v.x * nrm);
  unsafeAtomicAdd(od + 1, v.y * nrm);
  unsafeAtomicAdd(od + 2, v.z * nrm);
  unsafeAtomicAdd(od + 3, v.w * nrm);
}

// out = relu(out + b)
__global__ void k_bias_relu(float* __restrict__ out, const float* __restrict__ b,
                            long long total) {
  long long i = (long long)blockIdx.x * blockDim.x + threadIdx.x;
  if (i < total) {
    float v = out[i] + b[i & (D - 1)];
    out[i] = v > 0.0f ? v : 0.0f;
  }
}

// ---------------------------------------------------------------------------
extern "C" void kernel_launch(void* const* d_in, const int* in_sizes, int n_in,
                              void* d_out, int out_size, void* d_ws, size_t ws_size,
                              hipStream_t stream) {
  const float*     x  = (const float*)d_in[0];
  const long long* ei = (const long long*)d_in[1];
  const float*     ew = (const float*)d_in[2];
  const float*     W  = (const float*)d_in[3];
  const float*     b  = (const float*)d_in[4];
  float* out = (float*)d_out;

  const int n = in_sizes[0] / D;     // 50000
  const int E = in_sizes[2];         // 600000
  const long long total = (long long)n * D;

  // workspace partition (floats): [ Wt: D*D | dinv: n | h: n*D ]
  float* ws   = (float*)d_ws;
  float* Wt   = ws;
  float* dinv = ws + D * D;
  float* h    = ws + D * D + ((n + 3) & ~3);   // 16B-aligned for float4

  // 1) degree (init 1.0 for self-loop) + scatter + rsqrt
  k_init_deg<<<(n + 255) / 256, 256, 0, stream>>>(dinv, n);
  k_deg_scatter<<<(E + 255) / 256, 256, 0, stream>>>(ei, ew, dinv, E);
  k_dinv<<<(n + 255) / 256, 256, 0, stream>>>(dinv, n);

  // 2) W transpose, then h = x @ W via WMMA (4 waves/block, 1 row-tile/wave)
  k_transpose<<<(D * D + 255) / 256, 256, 0, stream>>>(W, Wt);
  const int numTiles = (n + 15) / 16;
  k_gemm_wmma<<<(numTiles + 3) / 4, 128, 0, stream>>>(x, Wt, h, n);

  // 3) out = self-loop term; 4) edge scatter atomics; 5) bias + relu
  k_selfloop<<<(int)((total + 255) / 256), 256, 0, stream>>>(h, dinv, out, total);
  long long sthreads = (long long)E * 32;
  k_edge_scatter<<<(int)((sthreads + 255) / 256), 256, 0, stream>>>(h, ei, ew, dinv, out, E);
  k_bias_relu<<<(int)((total + 255) / 256), 256, 0, stream>>>(out, b, total);
}